// UncertaintyTripletLoss_27006754357707
// MI455X (gfx1250) — compile-verified
//
#include <hip/hip_runtime.h>
#include <hip/hip_bf16.h>

// ---------------------------------------------------------------------------
// UncertaintyTripletLoss for MI455X (gfx1250, wave32).
// Memory-bound: 402 MB streamed (recon term) -> ~17us floor at 23.3 TB/s.
// V_WMMA_F32_16X16X4_F32 (ones-B trick) does wave reductions on the matrix
// pipe; block-level LDS stage (second WMMA) cuts same-address fp32 atomics
// 8x so atomic serialization stays off the HBM-roofline critical path.
// ---------------------------------------------------------------------------

typedef float v2f __attribute__((ext_vector_type(2)));
typedef float v4f __attribute__((ext_vector_type(4)));
typedef float v8f __attribute__((ext_vector_type(8)));

#define N_IMG   (1024 * 3 * 128 * 128)  // 50,331,648
#define N_LAT   (1024 * 128)            // 131,072
#define N_B     1024
#define N_D     128

// ds_swizzle SWAPX16: group-of-32 mode (offset[15]=0), xor=0x10, or=0, and=0x1F
// -> one DS op for the lane<->lane^16 pair swap, no index math.
__device__ __forceinline__ float swap16(float x) {
  return __int_as_float(__builtin_amdgcn_ds_swizzle(__float_as_int(x), 0x401F));
}

// Wave-wide sum of one float per lane on the matrix pipe.
// A = (x, 0) per lane, B = all-ones 4x16: D[m][n] = x_m + x_{m+16}.
// Lane n<16 holds D[0..7][n]; lane n>=16 holds D[8..15][n-16]. Summing the 8
// D components then the xor-16 swap yields the full 32-lane sum in every lane.
// Requires EXEC == all 1s (call from straight-line, non-divergent code).
__device__ __forceinline__ float wave_sum32(float x) {
  v2f a; a[0] = x;    a[1] = 0.0f;
  v2f b; b[0] = 1.0f; b[1] = 1.0f;
  v8f c = {};
  v8f d = __builtin_amdgcn_wmma_f32_16x16x4_f32(
      /*neg_a=*/false, a, /*neg_b=*/false, b,
      /*c_mod=*/(short)0, c, /*reuse_a=*/false, /*reuse_b=*/false);
  float s = ((d[0] + d[1]) + (d[2] + d[3])) + ((d[4] + d[5]) + (d[6] + d[7]));
  s += swap16(s);
  return s;
}

// Block-wide sum for 256-thread blocks (8 waves): wave WMMA reduce -> LDS ->
// wave 0 WMMA reduce. Returns total in wave 0 (valid in lane 0); one atomic
// per block. All waves must reach this with full EXEC.
__device__ __forceinline__ void block_sum_atomic(float x, float* dst) {
  __shared__ float red[8];
  float s = wave_sum32(x);
  int wid  = threadIdx.x >> 5;
  int lane = threadIdx.x & 31;
  if (lane == 0) red[wid] = s;
  __syncthreads();
  if (threadIdx.x < 32) {  // wave 0 fully active -> EXEC all 1s inside
    float v = (lane < 8) ? red[lane] : 0.0f;
    v = wave_sum32(v);
    if (lane == 0) atomicAdd(dst, v);
  }
}

__global__ void init_ws_kernel(float* __restrict__ ws) {
  int t = threadIdx.x;
  if (t < 3) ws[t] = 0.0f;
}

// recon_sum = sum( r*log(f) + (1-r)*log(1-f) )  (rewritten r*(lf-l1f)+l1f)
__global__ __launch_bounds__(256) void recon_kernel(
    const float* __restrict__ real, const float* __restrict__ fake,
    float* __restrict__ ws) {
  const int nv = N_IMG / 4;
  int tid = blockIdx.x * blockDim.x + threadIdx.x;
  int stride = gridDim.x * blockDim.x;
  const v4f* r4 = (const v4f*)real;
  const v4f* f4 = (const v4f*)fake;
  float acc = 0.0f;
  for (int i = tid; i < nv; i += stride) {
    v4f r = __builtin_nontemporal_load(&r4[i]);  // 402MB stream > 192MB L2
    v4f f = __builtin_nontemporal_load(&f4[i]);
#pragma unroll
    for (int k = 0; k < 4; ++k) {
      float lf  = __logf(f[k]);
      float l1f = __logf(1.0f - f[k]);
      acc = __fmaf_rn(r[k], lf - l1f, acc + l1f);
    }
  }
  block_sum_atomic(acc, &ws[0]);
}

// kl_sum = sum( 1 + logvar - mean^2 - exp(logvar) )
__global__ __launch_bounds__(256) void kl_kernel(
    const float* __restrict__ mean, const float* __restrict__ logvar,
    float* __restrict__ ws) {
  const int nv = N_LAT / 4;
  int tid = blockIdx.x * blockDim.x + threadIdx.x;
  int stride = gridDim.x * blockDim.x;
  const v4f* m4 = (const v4f*)mean;
  const v4f* l4 = (const v4f*)logvar;
  float acc = 0.0f;
  for (int i = tid; i < nv; i += stride) {
    v4f m = m4[i];
    v4f lv = l4[i];
#pragma unroll
    for (int k = 0; k < 4; ++k)
      acc += 1.0f + lv[k] - m[k] * m[k] - __expf(lv[k]);
  }
  block_sum_atomic(acc, &ws[1]);
}

// Per-dim JS for diagonal Gaussians with var_m = 0.25(va+vb):
//   (va+vb)/var_m == 4 identically; the two squared-mean terms collapse to
//   0.5*(ma-mb)^2/var_m, giving
//   js_dim = 0.25*(2*log(vm) - la - lb + 2 + 0.5*dm^2/vm).
// d = -sum(js); prob = softmax([d_ap,d_an])[0] = 1/(1+exp(S_ap - S_an)).
__device__ __forceinline__ float js_dim(float ma, float la, float va,
                                        float mb, float lb) {
  float vb = __expf(lb);
  float vm = 0.25f * (va + vb);
  float dm = ma - mb;
  return 0.25f * (2.0f * __logf(vm) - la - lb + 2.0f + 0.5f * dm * dm / vm);
}

__global__ __launch_bounds__(256) void triplet_kernel(
    const float* __restrict__ am, const float* __restrict__ alv,
    const float* __restrict__ pm, const float* __restrict__ plv,
    const float* __restrict__ nm, const float* __restrict__ nlv,
    float* __restrict__ ws) {
  __shared__ float red[8];
  int wave = (blockIdx.x * blockDim.x + threadIdx.x) >> 5;  // one wave per row
  int wid  = threadIdx.x >> 5;
  int lane = threadIdx.x & 31;
  const float* amr  = am  + wave * N_D;
  const float* alvr = alv + wave * N_D;
  const float* pmr  = pm  + wave * N_D;
  const float* plvr = plv + wave * N_D;
  const float* nmr  = nm  + wave * N_D;
  const float* nlvr = nlv + wave * N_D;
  float s_ap = 0.0f, s_an = 0.0f;
#pragma unroll
  for (int k = 0; k < N_D / 32; ++k) {
    int j = lane + k * 32;
    float ma = amr[j];
    float la = alvr[j];
    float va = __expf(la);
    s_ap += js_dim(ma, la, va, pmr[j], plvr[j]);
    s_an += js_dim(ma, la, va, nmr[j], nlvr[j]);
  }
  float S_ap = wave_sum32(s_ap);
  float S_an = wave_sum32(s_an);
  // d_an - d_ap = S_ap - S_an
  float prob = 1.0f / (1.0f + __expf(S_ap - S_an));
  if (lane == 0) red[wid] = prob;
  __syncthreads();
  if (threadIdx.x < 32) {  // wave 0 fully active
    float v = (lane < 8) ? red[lane] : 0.0f;
    v = wave_sum32(v);
    if (lane == 0) atomicAdd(&ws[2], v);
  }
}

__global__ void finalize_kernel(const float* __restrict__ ws,
                                float* __restrict__ out) {
  if (threadIdx.x == 0 && blockIdx.x == 0) {
    float recon = -ws[0] / (float)N_IMG;
    float kl    = -0.5f * ws[1] / (float)N_LAT;
    float trip  = -ws[2] / (float)N_B;
    out[0] = recon + kl + trip;  // final_loss (KL_BETA = TRIPLET_BETA = 1)
    out[1] = trip;               // triplet_error
    out[2] = recon;              // recon_error
    out[3] = kl;                 // kl_error
  }
}

extern "C" void kernel_launch(void* const* d_in, const int* in_sizes, int n_in,
                              void* d_out, int out_size, void* d_ws, size_t ws_size,
                              hipStream_t stream) {
  const float* real = (const float*)d_in[0];
  const float* fake = (const float*)d_in[1];
  const float* mean = (const float*)d_in[2];
  const float* lvar = (const float*)d_in[3];
  const float* am   = (const float*)d_in[4];
  const float* alv  = (const float*)d_in[5];
  const float* pm   = (const float*)d_in[6];
  const float* plv  = (const float*)d_in[7];
  const float* nm   = (const float*)d_in[8];
  const float* nlv  = (const float*)d_in[9];
  float* ws  = (float*)d_ws;
  float* out = (float*)d_out;
  (void)in_sizes; (void)n_in; (void)out_size; (void)ws_size;

  init_ws_kernel<<<1, 32, 0, stream>>>(ws);
  // 4096 blocks x 256 threads = 1,048,576 threads; 12 float4 iters each;
  // one fp32 atomic per block (4096 total, staggered -> off critical path).
  recon_kernel<<<4096, 256, 0, stream>>>(real, fake, ws);
  // 32768 vec4s over 32768 threads: one iteration each, 128 atomics.
  kl_kernel<<<128, 256, 0, stream>>>(mean, lvar, ws);
  // 1024 rows, one wave32 per row, 8 waves/block -> 128 blocks exactly.
  triplet_kernel<<<128, 256, 0, stream>>>(am, alv, pm, plv, nm, nlv, ws);
  finalize_kernel<<<1, 64, 0, stream>>>(ws, out);
}